// StrictGraphSphericalSNN_64123861729689
// MI455X (gfx1250) — compile-verified
//
#include <hip/hip_runtime.h>

// ---------------------------------------------------------------------------
// StrictGraphSphericalSNN for gfx1250 (MI455X)
//   - proj:  1x1 conv (3->16) via V_WMMA_F32_16X16X4_F32, bias folded into C
//   - lif:   9-point toroidal stencil avg + LIF recurrence, v[] in registers
//   - reductions: ballot/popcount + integer atomics (deterministic),
//                 fixed-order tree for float smoothness sum
// ---------------------------------------------------------------------------

#define HH 96
#define WW 96
#define NPIX (HH * WW)      // 9216
#define CIN 3
#define HID 16
#define OUTC 10
#define TT 8
#define BB 2
#define BETA 0.9f
#define THRESH 1.0f
#define NBLK2 ((BB * NPIX) / 256)   // 72 blocks in the LIF kernel

typedef __attribute__((ext_vector_type(2))) float v2f;
typedef __attribute__((ext_vector_type(8))) float v8f;

// ---------------------------------------------------------------------------
// Kernel 0: zero the accumulators (harness poisons ws once, never re-poisons)
// ---------------------------------------------------------------------------
__global__ void snn_init(int* __restrict__ spikecnt, float* __restrict__ smoothPart) {
    int tid = threadIdx.x;
    if (tid < TT * BB * HID) spikecnt[tid] = 0;
    if (tid < NBLK2) smoothPart[tid] = 0.0f;
}

// ---------------------------------------------------------------------------
// Kernel 1: z[t,b,p,h] = x[t,b,p,:] @ w_proj + b_proj  via WMMA f32 16x16x4
//   One wave per 16-pixel tile. K=4 (3 channels + zero pad).
//   A (16x4 f32): lanes 0-15 carry K={0,1}, lanes 16-31 carry K={2,pad}.
//   B (4x16 f32): mirrored layout of w_proj columns.
//   C: bias broadcast -> bias add is free.
// ---------------------------------------------------------------------------
__global__ __launch_bounds__(256) void snn_proj_wmma(
    const float* __restrict__ x,      // [T,B,NPIX,CIN]
    const float* __restrict__ wproj,  // [CIN,HID]
    const float* __restrict__ bproj,  // [HID]
    float* __restrict__ z)            // [T,B,NPIX,HID]
{
    const int wave = (blockIdx.x * blockDim.x + threadIdx.x) >> 5;
    const int lane = threadIdx.x & 31;
    const int tilesPerTB = NPIX / 16;            // 576
    const int tile = wave % tilesPerTB;
    const int tb   = wave / tilesPerTB;          // t*B + b, 0..15
    const int pbase = tile * 16;
    const int l15 = lane & 15;
    const bool hiHalf = (lane >= 16);

    // A-matrix: pixel row M = l15
    const float* xp = x + ((size_t)tb * NPIX + (pbase + l15)) * CIN;
    v2f a;
    if (!hiHalf) { a.x = xp[0]; a.y = xp[1]; }     // K=0, K=1
    else         { a.x = xp[2]; a.y = 0.0f; }      // K=2, K=3(pad)

    // B-matrix: column N = l15
    v2f bm;
    if (!hiHalf) { bm.x = wproj[0 * HID + l15]; bm.y = wproj[1 * HID + l15]; }
    else         { bm.x = wproj[2 * HID + l15]; bm.y = 0.0f; }

    // C-matrix: bias replicated down each column
    const float bias = bproj[l15];
    v8f c;
#pragma unroll
    for (int r = 0; r < 8; ++r) c[r] = bias;

    v8f d = __builtin_amdgcn_wmma_f32_16x16x4_f32(
        /*neg_a=*/false, a, /*neg_b=*/false, bm,
        /*c_mod=*/(short)0, c, /*reuse_a=*/false, /*reuse_b=*/false);

    // D layout: VGPR r -> (M=r, N=lane) lanes 0-15, (M=r+8, N=lane-16) lanes 16-31
    float* zt = z + (size_t)tb * NPIX * HID;
    const int prow = pbase + (hiHalf ? 8 : 0);
#pragma unroll
    for (int r = 0; r < 8; ++r)
        zt[(size_t)(prow + r) * HID + l15] = d[r];
}

// ---------------------------------------------------------------------------
// Kernel 2: stencil aggregation + LIF recurrence over T, per (b,pixel) thread.
//   Deterministic reductions:
//     - spikes: wave ballot -> popcount -> LDS int -> global int atomics
//     - smooth: per-thread accumulate over T, fixed-order LDS tree per block
// ---------------------------------------------------------------------------
__global__ __launch_bounds__(256) void snn_lif(
    const float* __restrict__ z,          // [T,B,NPIX,HID]
    int* __restrict__ spikecnt,           // [T,B,HID]
    float* __restrict__ smoothPart)       // [NBLK2]
{
    const int gid = blockIdx.x * blockDim.x + threadIdx.x;  // 0..B*NPIX-1
    const int b = gid / NPIX;                               // uniform per block
    const int p = gid % NPIX;
    const int y = p / WW, xc = p % WW;
    const int ym = (y + HH - 1) % HH, yp = (y + 1) % HH;
    const int xm = (xc + WW - 1) % WW, xp = (xc + 1) % WW;

    int nb[9];
    nb[0] = y  * WW + xc;   // self first (needed as the "flat" center)
    nb[1] = ym * WW + xc;  nb[2] = yp * WW + xc;
    nb[3] = y  * WW + xm;  nb[4] = y  * WW + xp;
    nb[5] = ym * WW + xm;  nb[6] = ym * WW + xp;
    nb[7] = yp * WW + xm;  nb[8] = yp * WW + xp;

    float v[HID];
#pragma unroll
    for (int h = 0; h < HID; ++h) v[h] = 0.0f;

    float smoothAcc = 0.0f;
    const float inv9 = 1.0f / 9.0f;

    __shared__ int bcnt[HID];
    __shared__ float sred[256];

    for (int t = 0; t < TT; ++t) {
        if (threadIdx.x < HID) bcnt[threadIdx.x] = 0;
        __syncthreads();

        const float* zbase = z + ((size_t)(t * BB + b) * NPIX) * HID;
        float agg[HID], ctr[HID];
#pragma unroll
        for (int h = 0; h < HID; ++h) agg[h] = 0.0f;
#pragma unroll
        for (int k = 0; k < 9; ++k) {
            const float* zr = zbase + (size_t)nb[k] * HID;
#pragma unroll
            for (int h = 0; h < HID; ++h) {
                float val = zr[h];
                agg[h] += val;
                if (k == 0) ctr[h] = val;
            }
        }

#pragma unroll
        for (int h = 0; h < HID; ++h) {
            float aval = agg[h] * inv9;
            smoothAcc += fabsf(aval - ctr[h]);
            float vn = BETA * v[h] + aval;
            int s = (vn - THRESH > 0.0f) ? 1 : 0;
            v[h] = vn - (s ? THRESH : 0.0f);
            unsigned long long m = __ballot(s);
            if ((threadIdx.x & 31) == 0)
                atomicAdd(&bcnt[h], (int)__popcll(m));
        }
        __syncthreads();

        if (threadIdx.x < HID)
            atomicAdd(&spikecnt[(t * BB + b) * HID + threadIdx.x], bcnt[threadIdx.x]);
        // (flush then re-zero next iter is done by the same threads -> safe)
        __syncthreads();
    }

    // block-level fixed-order tree reduction of the smoothness sum
    sred[threadIdx.x] = smoothAcc;
    __syncthreads();
#pragma unroll
    for (int s = 128; s > 0; s >>= 1) {
        if (threadIdx.x < (unsigned)s) sred[threadIdx.x] += sred[threadIdx.x + s];
        __syncthreads();
    }
    if (threadIdx.x == 0) smoothPart[blockIdx.x] = sred[0];
}

// ---------------------------------------------------------------------------
// Kernel 3: finalize (single block). Head matmul on spike means, time-mean
// readout, global spike-rate and smoothness scalars.
//   d_out layout: out[B,OUT] (20) | logits_seq[T,B,OUT] (160) | sr (1) | smooth (1)
// ---------------------------------------------------------------------------
__global__ __launch_bounds__(256) void snn_finalize(
    const int* __restrict__ spikecnt,      // [T,B,HID]
    const float* __restrict__ smoothPart,  // [NBLK2]
    const float* __restrict__ whead,       // [HID,OUT]
    const float* __restrict__ bhead,       // [OUT]
    float* __restrict__ out)
{
    __shared__ float logits[TT * BB * OUTC];  // 160
    const int tid = threadIdx.x;

    if (tid < TT * BB * OUTC) {
        const int o  = tid % OUTC;
        const int tb = tid / OUTC;           // t*B + b
        const int* cnt = spikecnt + tb * HID;
        float acc = bhead[o];
        const float invN = 1.0f / (float)NPIX;
#pragma unroll
        for (int h = 0; h < HID; ++h)
            acc += ((float)cnt[h] * invN) * whead[h * OUTC + o];
        logits[tid] = acc;
        out[BB * OUTC + tid] = acc;          // logits_seq
    }
    __syncthreads();

    if (tid < BB * OUTC) {                   // time-mean readout
        const int b = tid / OUTC, o = tid % OUTC;
        float m = 0.0f;
        for (int t = 0; t < TT; ++t) m += logits[(t * BB + b) * OUTC + o];
        out[tid] = m * (1.0f / (float)TT);
    }

    if (tid == 0) {
        const float denom = (float)TT * (float)BB * (float)NPIX * (float)HID;
        long long total = 0;
        for (int i = 0; i < TT * BB * HID; ++i) total += spikecnt[i];
        out[BB * OUTC + TT * BB * OUTC + 0] = (float)total / denom;  // spike rate
        float sm = 0.0f;
        for (int i = 0; i < NBLK2; ++i) sm += smoothPart[i];         // fixed order
        out[BB * OUTC + TT * BB * OUTC + 1] = sm / denom;            // smoothness
    }
}

// ---------------------------------------------------------------------------
extern "C" void kernel_launch(void* const* d_in, const int* in_sizes, int n_in,
                              void* d_out, int out_size, void* d_ws, size_t ws_size,
                              hipStream_t stream) {
    (void)in_sizes; (void)n_in; (void)out_size; (void)ws_size;
    const float* x     = (const float*)d_in[0];  // [T,B,H,W,CIN]
    const float* wproj = (const float*)d_in[1];  // [CIN,HID]
    const float* bproj = (const float*)d_in[2];  // [HID]
    const float* whead = (const float*)d_in[3];  // [HID,OUT]
    const float* bhead = (const float*)d_in[4];  // [OUT]

    // workspace carving
    float* z = (float*)d_ws;                                  // T*B*NPIX*HID f32 (~9.4 MB)
    const size_t zElems = (size_t)TT * BB * NPIX * HID;
    int* spikecnt = (int*)(z + zElems);                       // T*B*HID ints
    float* smoothPart = (float*)(spikecnt + TT * BB * HID);   // NBLK2 floats

    snn_init<<<1, 256, 0, stream>>>(spikecnt, smoothPart);

    // one wave per 16-pixel tile: T*B*(NPIX/16) = 9216 waves, 8 waves/block
    const int totalWaves = TT * BB * (NPIX / 16);
    snn_proj_wmma<<<totalWaves / 8, 256, 0, stream>>>(x, wproj, bproj, z);

    snn_lif<<<NBLK2, 256, 0, stream>>>(z, spikecnt, smoothPart);

    snn_finalize<<<1, 256, 0, stream>>>(spikecnt, smoothPart, whead, bhead,
                                        (float*)d_out);
}